// ComCo_19370302505372
// MI455X (gfx1250) — compile-verified
//
#include <hip/hip_runtime.h>
#include <hip/hip_bf16.h>

// Problem constants (from reference setup_inputs).
#define BROWS 512
#define LSEQ  512
#define FDIM  64
#define VBINS 20000
#define TROWS 513   // counts take values 0..512 inclusive

typedef __attribute__((ext_vector_type(2))) float v2f;
typedef __attribute__((ext_vector_type(8))) float v8f;

// ---------------------------------------------------------------------------
// Kernel 1: build LUT  T[x][f] = relu(x*W1 + b1) @ W2 + b2   for x = 0..512
// Real GEMM: H[513x64] @ W2[64x64] via V_WMMA_F32_16X16X4_F32 (fp32 WMMA).
// grid = 33 row-tiles, block = 128 (4 waves), wave w handles N-tile w.
// ---------------------------------------------------------------------------
__global__ __launch_bounds__(128) void ComCo_build_table(
    const float* __restrict__ W1,   // [64]  (shape (1,F))
    const float* __restrict__ b1,   // [64]
    const float* __restrict__ W2,   // [64,64]
    const float* __restrict__ b2,   // [64]
    float* __restrict__ T) {        // [513,64]
  const int lane    = threadIdx.x & 31;
  const int wave    = threadIdx.x >> 5;   // N tile 0..3
  const int rowTile = blockIdx.x;         // M tile 0..32
  const int M       = lane & 15;          // row-in-tile (A) / col (B,D)
  const int khalf   = lane >> 4;          // 0: K{0,1} / M 0..7 ; 1: K{2,3} / M 8..15
  const int ncol    = wave * 16 + M;      // output column f
  const float x     = (float)(rowTile * 16 + M);  // A-matrix row value (count)

  v8f acc = {};
  #pragma unroll
  for (int k0 = 0; k0 < FDIM; k0 += 4) {
    const int ka = k0 + khalf * 2;
    // A fragment: A[M][ka], A[M][ka+1]  (hidden = relu(x*W1+b1))
    float h0 = x * W1[ka]     + b1[ka];
    float h1 = x * W1[ka + 1] + b1[ka + 1];
    v2f a, b;
    a.x = h0 > 0.0f ? h0 : 0.0f;
    a.y = h1 > 0.0f ? h1 : 0.0f;
    // B fragment: W2[ka][ncol], W2[ka+1][ncol]
    b.x = W2[ka * FDIM + ncol];
    b.y = W2[(ka + 1) * FDIM + ncol];
    acc = __builtin_amdgcn_wmma_f32_16x16x4_f32(
        /*neg_a=*/false, a, /*neg_b=*/false, b,
        /*c_mod=*/(short)0, acc, /*reuse_a=*/false, /*reuse_b=*/false);
  }
  // D layout: VGPR r -> M = r + 8*khalf, N = lane&15
  #pragma unroll
  for (int r = 0; r < 8; ++r) {
    const int row = rowTile * 16 + r + khalf * 8;
    if (row < TROWS) T[row * FDIM + ncol] = acc[r] + b2[ncol];
  }
}

// ---------------------------------------------------------------------------
// Kernel 2: per-row packed LDS histogram + LUT gather + coalesced stores.
// One block per row b. Packed hist: low16 = src count, high16 = dst count
// (counts <= 512, no carry). 80 KB LDS fits CDNA5's 320 KB/WGP.
// ---------------------------------------------------------------------------
__global__ __launch_bounds__(256) void ComCo_count_encode(
    const int* __restrict__ src_ids, const int* __restrict__ dst_ids,
    const float* __restrict__ T,
    float* __restrict__ out_src, float* __restrict__ out_dst) {
  __shared__ unsigned int hist[VBINS];     // 80000 B
  __shared__ unsigned int csrc[LSEQ];      // packed counts per position
  __shared__ unsigned int cdst[LSEQ];

  const int b = blockIdx.x;
  const int t = threadIdx.x;

  for (int i = t; i < VBINS; i += 256) hist[i] = 0u;
  __syncthreads();

  const int base = b * LSEQ;
  const int s0 = src_ids[base + t];
  const int s1 = src_ids[base + 256 + t];
  const int d0 = dst_ids[base + t];
  const int d1 = dst_ids[base + 256 + t];
  atomicAdd(&hist[s0], 1u);        // src hist in low 16 bits
  atomicAdd(&hist[s1], 1u);
  atomicAdd(&hist[d0], 65536u);    // dst hist in high 16 bits
  atomicAdd(&hist[d1], 65536u);
  __syncthreads();

  const unsigned ps0 = hist[s0], ps1 = hist[s1];
  const unsigned pd0 = hist[d0], pd1 = hist[d1];
  // id 0 is padding -> both count channels zero (maps to T[0], as reference)
  csrc[t]       = (s0 == 0) ? 0u : ps0;
  csrc[t + 256] = (s1 == 0) ? 0u : ps1;
  cdst[t]       = (d0 == 0) ? 0u : pd0;
  cdst[t + 256] = (d1 == 0) ? 0u : pd1;
  __syncthreads();

  // Emit: per row-block 512 positions x 16 float4 = 8192 float4 per tensor.
  // 256 threads -> 32 iterations; each wave stores 512 contiguous bytes.
  float4* osrc = (float4*)(out_src + (size_t)base * FDIM);
  float4* odst = (float4*)(out_dst + (size_t)base * FDIM);
  #pragma unroll 4
  for (int i = 0; i < 32; ++i) {
    const int fi = i * 256 + t;          // float4 index 0..8191
    const int l  = fi >> 4;              // position
    const int f  = (fi & 15) * 4;        // feature offset
    const unsigned c  = csrc[l];
    const unsigned c0 = c & 0xffffu;
    const unsigned c1 = c >> 16;
    const float4 v0 = *(const float4*)&T[c0 * FDIM + f];
    const float4 v1 = *(const float4*)&T[c1 * FDIM + f];
    float4 v;
    v.x = v0.x + v1.x; v.y = v0.y + v1.y;
    v.z = v0.z + v1.z; v.w = v0.w + v1.w;
    osrc[fi] = v;
  }
  #pragma unroll 4
  for (int i = 0; i < 32; ++i) {
    const int fi = i * 256 + t;
    const int l  = fi >> 4;
    const int f  = (fi & 15) * 4;
    const unsigned c  = cdst[l];
    const unsigned c0 = c & 0xffffu;
    const unsigned c1 = c >> 16;
    const float4 v0 = *(const float4*)&T[c0 * FDIM + f];
    const float4 v1 = *(const float4*)&T[c1 * FDIM + f];
    float4 v;
    v.x = v0.x + v1.x; v.y = v0.y + v1.y;
    v.z = v0.z + v1.z; v.w = v0.w + v1.w;
    odst[fi] = v;
  }
}

extern "C" void kernel_launch(void* const* d_in, const int* in_sizes, int n_in,
                              void* d_out, int out_size, void* d_ws, size_t ws_size,
                              hipStream_t stream) {
  const int*   src_ids = (const int*)d_in[0];
  const int*   dst_ids = (const int*)d_in[1];
  const float* W1      = (const float*)d_in[2];
  const float* b1      = (const float*)d_in[3];
  const float* W2      = (const float*)d_in[4];
  const float* b2      = (const float*)d_in[5];
  // d_in[6] = num_nodes (scalar) -- fixed at 20000 for this problem.

  float* T       = (float*)d_ws;                       // 513*64 floats = 131 KB
  float* out_src = (float*)d_out;                      // [512,512,64]
  float* out_dst = out_src + (size_t)BROWS * LSEQ * FDIM;

  ComCo_build_table<<<dim3((TROWS + 15) / 16), dim3(128), 0, stream>>>(
      W1, b1, W2, b2, T);
  ComCo_count_encode<<<dim3(BROWS), dim3(256), 0, stream>>>(
      src_ids, dst_ids, T, out_src, out_dst);
}